// RecurrentNesterovAttention_38852274160140
// MI455X (gfx1250) — compile-verified
//
#include <hip/hip_runtime.h>

typedef __attribute__((ext_vector_type(2))) float v2f;
typedef __attribute__((ext_vector_type(8))) float v8f;

#define NTOK 16384          // N*H = 1024*16
#define DDIM 64
#define MDIM 64
#define WPB  8              // waves per block (wave32)

__device__ __forceinline__ float elu1(float x) {
    // elu(x) + 1  ==  x>0 ? x+1 : exp(x)
    return x > 0.0f ? x + 1.0f : __expf(x);
}

__global__ __launch_bounds__(256, 1) void rna_kernel(
    const float* __restrict__ q,  const float* __restrict__ k,
    const float* __restrict__ val, const float* __restrict__ Si,
    const float* __restrict__ Zi, const float* __restrict__ Pi,
    const int*  __restrict__ tptr,
    float* __restrict__ outV, float* __restrict__ outS,
    float* __restrict__ outZ, float* __restrict__ outP)
{
    __shared__ float ldsQ[WPB][DDIM];
    __shared__ float ldsK[WPB][DDIM];

    const int tid  = threadIdx.x;
    const int wid  = tid >> 5;
    const int lane = tid & 31;
    const int nh   = blockIdx.x * WPB + wid;      // one wave == one (n,h)

    const int   t  = tptr[0] + 1;
    const float mu = (float)(t - 1) / (float)(t + 2);

    const size_t vbase = (size_t)nh * DDIM;
    const size_t mbase = (size_t)nh * (DDIM * MDIM);

    // ---- feature maps: lane covers d = lane and d = lane+32 ----
    const float qf0 = elu1(q[vbase + lane]);
    const float qf1 = elu1(q[vbase + lane + 32]);
    const float kf0 = elu1(k[vbase + lane]);
    const float kf1 = elu1(k[vbase + lane + 32]);
    ldsQ[wid][lane]      = qf0;  ldsQ[wid][lane + 32] = qf1;
    ldsK[wid][lane]      = kf0;  ldsK[wid][lane + 32] = kf1;

    // ---- Zi_new and normalizer Z = 1/(Q . Zi_new + eps) ----
    const float zn0 = Zi[vbase + lane]      + kf0;
    const float zn1 = Zi[vbase + lane + 32] + kf1;
    outZ[vbase + lane]      = zn0;
    outZ[vbase + lane + 32] = zn1;

    float zp = qf0 * zn0 + qf1 * zn1;
    #pragma unroll
    for (int off = 16; off > 0; off >>= 1)
        zp += __shfl_xor(zp, off, 32);
    const float zinv = 1.0f / (zp + 1e-6f);

    const int col  = lane & 15;   // N index inside a 16x16 tile
    const int half = lane >> 4;   // 0: rows v..v ; 1: rows v+8

    // value chunks for the B operand (only lanes 0-15 / K=0 row are used)
    float vb[4];
    #pragma unroll
    for (int c = 0; c < 4; ++c)
        vb[c] = val[vbase + 16 * c + col];

    float acc[4] = {0.f, 0.f, 0.f, 0.f};

    // ---- 4x4 grid of 16x16 tiles over the (D,M) state ----
    #pragma unroll
    for (int R = 0; R < 4; ++R) {
        // A (16x4 f32): K=0 column = 0.1*Kf chunk (lanes 0-15 of VGPR0), rest 0
        v2f A;
        A.x = (half == 0) ? 0.1f * ldsK[wid][16 * R + col] : 0.0f;
        A.y = 0.0f;

        // Q broadcast values for this lane's 8 rows of the tile
        float qr[8];
        #pragma unroll
        for (int vv = 0; vv < 8; ++vv)
            qr[vv] = ldsQ[wid][16 * R + 8 * half + vv];

        const size_t rowoff = mbase + (size_t)(16 * R + 8 * half) * MDIM + col;
        const float* siP = Si  + rowoff;
        const float* piP = Pi  + rowoff;
        float*       poP = outP + rowoff;
        float*       soP = outS + rowoff;

        #pragma unroll
        for (int c = 0; c < 4; ++c) {
            // B (4x16 f32): K=0 row = value chunk, rest 0
            v2f B;
            B.x = (half == 0) ? vb[c] : 0.0f;
            B.y = 0.0f;

            const int o = 16 * c;

            // Si tile loaded directly in native WMMA C layout (b32, coalesced)
            v8f C;
            #pragma unroll
            for (int vv = 0; vv < 8; ++vv)
                C[vv] = __builtin_nontemporal_load(siP + o + vv * MDIM);

            // Pinext tile = Si + 0.1 * K (x) V   — fused on the matrix pipe
            v8f Dm = __builtin_amdgcn_wmma_f32_16x16x4_f32(
                false, A, false, B, (short)0, C, false, false);

            #pragma unroll
            for (int vv = 0; vv < 8; ++vv) {
                const float pi_ = __builtin_nontemporal_load(piP + o + vv * MDIM);
                const float pn  = Dm[vv];
                const float sn  = fmaf(mu, pn - pi_, pn);   // Nesterov blend
                __builtin_nontemporal_store(pn, poP + o + vv * MDIM);
                __builtin_nontemporal_store(sn, soP + o + vv * MDIM);
                acc[c] = fmaf(qr[vv], sn, acc[c]);          // V reduction
            }
        }
    }

    // combine the two half-wave row sets (rows v vs v+8)
    #pragma unroll
    for (int c = 0; c < 4; ++c)
        acc[c] += __shfl_xor(acc[c], 16, 32);

    if (half == 0) {
        #pragma unroll
        for (int c = 0; c < 4; ++c)
            outV[vbase + 16 * c + col] = acc[c] * zinv;
    }
}

extern "C" void kernel_launch(void* const* d_in, const int* in_sizes, int n_in,
                              void* d_out, int out_size, void* d_ws, size_t ws_size,
                              hipStream_t stream) {
    (void)in_sizes; (void)n_in; (void)out_size; (void)d_ws; (void)ws_size;

    const float* q   = (const float*)d_in[0];
    const float* k   = (const float*)d_in[1];
    const float* v   = (const float*)d_in[2];
    const float* Si  = (const float*)d_in[3];
    const float* Zi  = (const float*)d_in[4];
    const float* Pi  = (const float*)d_in[5];
    const int*   ti  = (const int*)d_in[6];

    // outputs concatenated in return order: V, Si_new, Zi_new, Pinext
    float* outV = (float*)d_out;
    float* outS = outV + (size_t)NTOK * MDIM;
    float* outZ = outS + (size_t)NTOK * DDIM * MDIM;
    float* outP = outZ + (size_t)NTOK * DDIM;

    rna_kernel<<<dim3(NTOK / WPB), dim3(256), 0, stream>>>(
        q, k, v, Si, Zi, Pi, ti, outV, outS, outZ, outP);
}